// HierarchicalLoss_21131239096524
// MI455X (gfx1250) — compile-verified
//
#include <hip/hip_runtime.h>
#include <hip/hip_bf16.h>
#include <math.h>

// ---------------------------------------------------------------------------
// HierarchicalLoss for MI455X (gfx1250, wave32).
//   B=8, S=4096, V=256, D=512, stride=2, SPACE=63, TEMP=0.07
//   contrastive_2: (8,128,1024) -> N2=4096 rows of dim 128
//   contrastive_3: (8,128,256)  -> N3=1024 rows of dim 128
// InfoNCE GEMMs use V_WMMA_F32_16X16X4_F32 fused with row-wise sum(exp)
// (no max-shift needed: |logit| <= 1/0.07 ~ 14.3, exp sums < 7e9 in fp32).
// Two-phase LSE: column-chunked partial exp-sums -> per-row log.
// Workspace requirement: ~5.4 MB.
// ---------------------------------------------------------------------------

typedef float v2f __attribute__((ext_vector_type(2)));
typedef float v8f __attribute__((ext_vector_type(8)));

#define SPACE_TOK 63
#define N_B 8
#define N_S 4096
#define N_V 256
#define N_D 512
#define N_WLEN 2048
#define DIM_C 128
#define INV_TEMP (1.0f / 0.07f)

// acc layout (floats): 0=sum(ce*mask) 1=sum(mask) 2=sum(ce)
//                      3=word tot     4=word cnt  5=nce2 sum(lse-diag) 6=nce3

// --------------------------- char cross-entropy ----------------------------
__global__ void ce_kernel(const float* __restrict__ logits,
                          const int* __restrict__ targets,
                          const float* __restrict__ mask,
                          float* __restrict__ acc) {
  int wid = threadIdx.x >> 5;
  int lane = threadIdx.x & 31;
  int row = blockIdx.x * 8 + wid;            // 0 .. B*S-1
  const float* x = logits + (size_t)row * N_V;

  float v[8];
  float m = -3.0e38f;
#pragma unroll
  for (int q = 0; q < 8; ++q) {
    v[q] = x[lane + q * 32];
    m = fmaxf(m, v[q]);
  }
#pragma unroll
  for (int off = 16; off > 0; off >>= 1) m = fmaxf(m, __shfl_xor(m, off, 32));
  float s = 0.0f;
#pragma unroll
  for (int q = 0; q < 8; ++q) s += expf(v[q] - m);
#pragma unroll
  for (int off = 16; off > 0; off >>= 1) s += __shfl_xor(s, off, 32);
  float lse = m + logf(s);

  __shared__ float sdata[8][3];
  if (lane == 0) {
    int t = targets[row];
    float ce = lse - x[t];
    float mk = mask[row];
    sdata[wid][0] = ce * mk;
    sdata[wid][1] = mk;
    sdata[wid][2] = ce;
  }
  __syncthreads();
  if (threadIdx.x == 0) {
    float a = 0.0f, b = 0.0f, c = 0.0f;
#pragma unroll
    for (int w = 0; w < 8; ++w) { a += sdata[w][0]; b += sdata[w][1]; c += sdata[w][2]; }
    atomicAdd(acc + 0, a);
    atomicAdd(acc + 1, b);
    atomicAdd(acc + 2, c);
  }
}

// --------------------- word span loss: find space positions ----------------
__global__ void find_spaces(const int* __restrict__ targets,
                            int* __restrict__ nsp, int* __restrict__ spos) {
  int b = blockIdx.x;
  int tid = threadIdx.x;                     // 256 threads, 16 positions each
  const int* t = targets + (size_t)b * N_S;
  __shared__ int cnt[257];
  int c = 0;
  for (int q = 0; q < 16; ++q)
    if (t[tid * 16 + q] == SPACE_TOK) ++c;
  cnt[tid + 1] = c;
  __syncthreads();
  if (tid == 0) {
    cnt[0] = 0;
    for (int i = 1; i <= 256; ++i) cnt[i] += cnt[i - 1];
    nsp[b] = cnt[256];
  }
  __syncthreads();
  int o = cnt[tid];
  int* sp = spos + (size_t)b * N_S;
  for (int q = 0; q < 16; ++q) {
    int p = tid * 16 + q;
    if (t[p] == SPACE_TOK) sp[o++] = p;
  }
}

// --------------------- word span loss: per-segment MSE ----------------------
__global__ void word_seg(const float* __restrict__ feats,
                         const float* __restrict__ wpred,
                         const int* __restrict__ nsp,
                         const int* __restrict__ spos,
                         float* __restrict__ acc) {
  int b = blockIdx.y;
  int k = blockIdx.x + 1;                    // segment id, 1..n_sp-1
  int ns = nsp[b];
  if (k > ns - 1) return;
  const int* sp = spos + (size_t)b * N_S;
  int p0 = sp[k - 1];                        // k-th space position
  int p1 = sp[k];                            // (k+1)-th space position
  int len = p1 - p0 - 1;
  int wp = p0 >> 1;                          // pos // stride
  if (len <= 0 || wp >= N_WLEN) return;

  int tid = threadIdx.x;                     // 128 threads
  float local = 0.0f;
  float invLen = 1.0f / (float)len;
  for (int d = tid; d < N_D; d += 128) {
    const float* f = feats + ((size_t)b * N_D + d) * N_S;
    float sum = 0.0f;
    for (int t = p0 + 1; t < p1; ++t) sum += f[t];
    float mean = sum * invLen;
    float pred = wpred[((size_t)b * N_D + d) * N_WLEN + wp];
    float diff = pred - mean;
    local += diff * diff;
  }
  __shared__ float red[128];
  red[tid] = local;
  __syncthreads();
  for (int s = 64; s > 0; s >>= 1) {
    if (tid < s) red[tid] += red[tid + s];
    __syncthreads();
  }
  if (tid == 0) {
    atomicAdd(acc + 3, red[0] * (1.0f / (float)N_D));
    atomicAdd(acc + 4, 1.0f);
  }
}

// ------------------- InfoNCE: normalize + pack row-major --------------------
// emb: (B, 128, L). Row i = b*npairs + j; a = emb[b,:,2j], p = emb[b,:,2j+1].
__global__ void normalize_pairs(const float* __restrict__ emb,
                                float* __restrict__ Aw, float* __restrict__ Pw,
                                int L, int npairs) {
  int i = blockIdx.x;
  int d = threadIdx.x;                       // 128 threads == DIM_C
  int b = i / npairs;
  int j = i - b * npairs;
  size_t off = ((size_t)b * DIM_C + d) * L;
  float av = emb[off + 2 * j];
  float pv = emb[off + 2 * j + 1];
  __shared__ float ra[128], rp[128];
  ra[d] = av * av;
  rp[d] = pv * pv;
  __syncthreads();
  for (int s = 64; s > 0; s >>= 1) {
    if (d < s) { ra[d] += ra[d + s]; rp[d] += rp[d + s]; }
    __syncthreads();
  }
  __shared__ float na, npv;
  if (d == 0) {
    na = fmaxf(sqrtf(ra[0]), 1e-12f);
    npv = fmaxf(sqrtf(rp[0]), 1e-12f);
  }
  __syncthreads();
  Aw[(size_t)i * DIM_C + d] = av / na;
  Pw[(size_t)i * DIM_C + d] = pv / npv;
}

// -------- InfoNCE phase A: WMMA GEMM + partial row exp-sums + diag ---------
// Block = 128 threads = 4 waves; each wave owns 16 rows (A panel in regs),
// P tiles (16 x 128) streamed through LDS (row stride 132 -> conflict-free).
// Grid = (rowBlocks, colChunks); each block covers column tiles
// [blockIdx.y*ntc, blockIdx.y*ntc + ntc). Partial exp-sums are atomically
// added into rowsum[]; -diag goes straight into the loss slot.
__global__ void nce_partial(const float* __restrict__ Aw,
                            const float* __restrict__ Pw,
                            int ntc,                        // tiles per chunk
                            float* __restrict__ rowsum,
                            float* __restrict__ acc_slot) {
  __shared__ float sP[16 * 132];
  int tid = threadIdx.x;
  int wid = tid >> 5;
  int lane = tid & 31;
  int ib = blockIdx.x * 64 + wid * 16;       // wave's first row
  int m = lane & 15;                         // matrix row (A) / col (B)
  int khalf = (lane >> 4) << 1;              // 0 for lanes 0-15, 2 for 16-31
  int tile0 = blockIdx.y * ntc;

  // Preload this wave's 16x128 A panel as 32 K-step fragments (ISA layout:
  // VGPR0 = K+khalf, VGPR1 = K+khalf+1).
  const float* arow = Aw + (size_t)(ib + m) * DIM_C;
  v2f af[32];
#pragma unroll
  for (int kk = 0; kk < 32; ++kk) {
    af[kk].x = arow[kk * 4 + khalf];
    af[kk].y = arow[kk * 4 + khalf + 1];
  }

  float esum[8] = {0.f, 0.f, 0.f, 0.f, 0.f, 0.f, 0.f, 0.f};
  float diag = 0.0f;

  for (int jj = 0; jj < ntc; ++jj) {
    int jt = tile0 + jj;
    __syncthreads();
    for (int idx = tid; idx < 16 * 128; idx += 128) {
      int r = idx >> 7;
      int cpos = idx & 127;
      sP[r * 132 + cpos] = Pw[(size_t)(jt * 16 + r) * DIM_C + cpos];
    }
    __syncthreads();
    if (jj + 1 < ntc) {
      // pull next P tile (8 KB) toward the WGP while WMMAs run
      __builtin_prefetch(Pw + (size_t)(jt + 1) * 16 * DIM_C + tid * 16, 0, 0);
    }

    // Two independent accumulator chains over K (halves WMMA RAW depth).
    v8f c0 = {0.f, 0.f, 0.f, 0.f, 0.f, 0.f, 0.f, 0.f};
    v8f c1 = {0.f, 0.f, 0.f, 0.f, 0.f, 0.f, 0.f, 0.f};
#pragma unroll
    for (int kk = 0; kk < 16; ++kk) {
      v2f bf0, bf1;
      bf0.x = sP[m * 132 + kk * 4 + khalf];
      bf0.y = sP[m * 132 + kk * 4 + khalf + 1];
      bf1.x = sP[m * 132 + (kk + 16) * 4 + khalf];
      bf1.y = sP[m * 132 + (kk + 16) * 4 + khalf + 1];
      c0 = __builtin_amdgcn_wmma_f32_16x16x4_f32(
          false, af[kk], false, bf0, (short)0, c0, false, false);
      c1 = __builtin_amdgcn_wmma_f32_16x16x4_f32(
          false, af[kk + 16], false, bf1, (short)0, c1, false, false);
    }

    bool dtile = (jt * 16 == ib);
#pragma unroll
    for (int r = 0; r < 8; ++r) {
      float val = (c0[r] + c1[r]) * INV_TEMP;  // logit
      esum[r] += expf(val);                    // |val| <= 14.3 -> safe
      if (dtile) {
        int mrow = r + ((lane >= 16) ? 8 : 0);
        if ((lane & 15) == mrow) diag += val;
      }
    }
  }

  // Row sums live across each 16-lane half (one column per lane).
#pragma unroll
  for (int r = 0; r < 8; ++r) {
#pragma unroll
    for (int off = 1; off <= 8; off <<= 1)
      esum[r] += __shfl_xor(esum[r], off, 32);
  }
  if (lane == 0) {
#pragma unroll
    for (int r = 0; r < 8; ++r) atomicAdd(rowsum + ib + r, esum[r]);
  } else if (lane == 16) {
#pragma unroll
    for (int r = 0; r < 8; ++r) atomicAdd(rowsum + ib + 8 + r, esum[r]);
  }
#pragma unroll
  for (int off = 1; off <= 16; off <<= 1) diag += __shfl_xor(diag, off, 32);
  if (lane == 0 && diag != 0.0f) atomicAdd(acc_slot, -diag);
}

// -------- InfoNCE phase B: lse_i = log(rowsum_i), accumulate sum ----------
__global__ void nce_lse(const float* __restrict__ rowsum, int N,
                        float* __restrict__ acc_slot) {
  int i = blockIdx.x * 256 + threadIdx.x;
  float v = (i < N) ? logf(rowsum[i]) : 0.0f;
  __shared__ float red[256];
  red[threadIdx.x] = v;
  __syncthreads();
  for (int s = 128; s > 0; s >>= 1) {
    if (threadIdx.x < s) red[threadIdx.x] += red[threadIdx.x + s];
    __syncthreads();
  }
  if (threadIdx.x == 0) atomicAdd(acc_slot, red[0]);
}

// ------------------------------- finalize ----------------------------------
__global__ void finalize(const float* __restrict__ acc, float* __restrict__ out) {
  float scm = acc[0], sm = acc[1], sce = acc[2];
  float wtot = acc[3], wcnt = acc[4];
  float a2 = acc[5], a3 = acc[6];
  float char_loss = (sm > 0.0f) ? scm / fmaxf(sm, 1.0f)
                                : sce / (float)(N_B * N_S);
  float word_loss = (wcnt > 0.0f) ? wtot / fmaxf(wcnt, 1.0f) : 0.0f;
  float phrase = a2 / 4096.0f;               // mean over N2 rows
  float idea = a3 / 1024.0f;                 // mean over N3 rows
  out[0] = 1.0f * char_loss + 0.5f * word_loss + 0.1f * phrase + 0.1f * idea;
}

// ------------------------------- launcher ----------------------------------
extern "C" void kernel_launch(void* const* d_in, const int* in_sizes, int n_in,
                              void* d_out, int out_size, void* d_ws, size_t ws_size,
                              hipStream_t stream) {
  const float* logits = (const float*)d_in[0];        // (8,4096,256)
  const float* word_preds = (const float*)d_in[1];    // (8,512,2048)
  const float* char_features = (const float*)d_in[2]; // (8,512,4096)
  const float* c2 = (const float*)d_in[3];            // (8,128,1024)
  const float* c3 = (const float*)d_in[4];            // (8,128,256)
  const int* targets = (const int*)d_in[5];           // (8,4096)
  const float* mask = (const float*)d_in[6];          // (8,4096)
  float* out = (float*)d_out;

  // workspace layout (all zeroed floats first, then scratch)
  float* acc = (float*)d_ws;                 // 16 floats
  float* rowsum2 = acc + 16;                 // 4096
  float* rowsum3 = rowsum2 + 4096;           // 1024
  int* nsp = (int*)(rowsum3 + 1024);         // 16 ints (8 used)
  int* spos = nsp + 16;                      // 8*4096 ints
  float* A2 = (float*)(spos + N_B * N_S);    // 4096*128
  float* P2 = A2 + 4096 * DIM_C;             // 4096*128
  float* A3 = P2 + 4096 * DIM_C;             // 1024*128
  float* P3 = A3 + 1024 * DIM_C;             // 1024*128
  (void)in_sizes; (void)n_in; (void)out_size; (void)ws_size;

  hipMemsetAsync(acc, 0, (16 + 4096 + 1024) * sizeof(float), stream);

  // 1. char CE: one wave per (b,s) row
  ce_kernel<<<N_B * N_S / 8, 256, 0, stream>>>(logits, targets, mask, acc);

  // 2. word span loss
  find_spaces<<<N_B, 256, 0, stream>>>(targets, nsp, spos);
  word_seg<<<dim3(N_S - 1, N_B), 128, 0, stream>>>(char_features, word_preds,
                                                   nsp, spos, acc);

  // 3/4. InfoNCE: normalize, then fused WMMA GEMM + two-phase LSE
  normalize_pairs<<<4096, 128, 0, stream>>>(c2, A2, P2, 1024, 512);
  normalize_pairs<<<1024, 128, 0, stream>>>(c3, A3, P3, 256, 128);
  // c2: 4096x4096, 64 row-blocks x 8 column chunks (32 tiles each) = 512 blocks
  nce_partial<<<dim3(64, 8), 128, 0, stream>>>(A2, P2, 32, rowsum2, acc + 5);
  // c3: 1024x1024, 16 row-blocks x 4 column chunks (16 tiles each) = 64 blocks
  nce_partial<<<dim3(16, 4), 128, 0, stream>>>(A3, P3, 16, rowsum3, acc + 6);
  nce_lse<<<4096 / 256, 256, 0, stream>>>(rowsum2, 4096, acc + 5);
  nce_lse<<<1024 / 256, 256, 0, stream>>>(rowsum3, 1024, acc + 6);

  // 5. combine
  finalize<<<1, 1, 0, stream>>>(acc, out);
}